// StructuralLayerHyperRec_74861279969936
// MI455X (gfx1250) — compile-verified
//
#include <hip/hip_runtime.h>
#include <hip/hip_bf16.h>

#define T_STEPS 3
#define NU 100000
#define NV 50000
#define NE 50000
#define DD 128
#define NNZ 800000
#define NHGCN 2
#define NLG 2

typedef __attribute__((ext_vector_type(2))) float v2f;
typedef __attribute__((ext_vector_type(8))) float v8f;

static __host__ int cdiv(long long a, long long b) { return (int)((a + b - 1) / b); }

// ---------------------------------------------------------------- elementwise
__global__ void k_zero(float* __restrict__ p, long long n4) {
    long long g = (long long)blockIdx.x * blockDim.x + threadIdx.x;
    if (g < n4) *(float4*)(p + g * 4) = make_float4(0.f, 0.f, 0.f, 0.f);
}

__global__ void k_relu(float* __restrict__ p, long long n4) {
    long long g = (long long)blockIdx.x * blockDim.x + threadIdx.x;
    if (g >= n4) return;
    float4 v = *(float4*)(p + g * 4);
    v.x = fmaxf(v.x, 0.f); v.y = fmaxf(v.y, 0.f);
    v.z = fmaxf(v.z, 0.f); v.w = fmaxf(v.w, 0.f);
    *(float4*)(p + g * 4) = v;
}

__global__ void k_copy(float* __restrict__ d, const float* __restrict__ s, long long n4) {
    long long g = (long long)blockIdx.x * blockDim.x + threadIdx.x;
    if (g < n4) *(float4*)(d + g * 4) = *(const float4*)(s + g * 4);
}

__global__ void k_add(float* __restrict__ d, const float* __restrict__ s, long long n4) {
    long long g = (long long)blockIdx.x * blockDim.x + threadIdx.x;
    if (g >= n4) return;
    float4 a = *(float4*)(d + g * 4);
    float4 b = *(const float4*)(s + g * 4);
    a.x += b.x; a.y += b.y; a.z += b.z; a.w += b.w;
    *(float4*)(d + g * 4) = a;
}

// --------------------------------------------------------------- gathers
// sf[n] = static[gv[n]] ; df[n] = dyn_v[trace_v[gv[n]*T + t]]
__global__ void k_gather_sfdf(const float* __restrict__ stat, const int* __restrict__ gv,
                              const int* __restrict__ trace_v, const float* __restrict__ xpool,
                              float* __restrict__ sf, float* __restrict__ df, int t) {
    long long g = (long long)blockIdx.x * blockDim.x + threadIdx.x;
    int n = (int)(g >> 5);
    int c = (int)(g & 31) << 2;
    if (n >= NV) return;
    int gi = gv[n];
    *(float4*)(sf + (size_t)n * DD + c) = *(const float4*)(stat + (size_t)gi * DD + c);
    int tv = trace_v[(size_t)gi * T_STEPS + t];
    const float* src = (tv < NU) ? (stat + (size_t)tv * DD) : (xpool + (size_t)(tv - NU) * DD);
    *(float4*)(df + (size_t)n * DD + c) = *(const float4*)(src + c);
}

// out[u*T+t] = (idx<NU) ? static[idx] : pool[idx-NU]
__global__ void k_final_gather(const int* __restrict__ trace, const float* __restrict__ stat,
                               const float* __restrict__ pool, float* __restrict__ out) {
    long long g = (long long)blockIdx.x * blockDim.x + threadIdx.x;
    int ut = (int)(g >> 5);
    int c = (int)(g & 31) << 2;
    if (ut >= NU * T_STEPS) return;
    int idx = trace[ut];
    const float* src = (idx < NU) ? (stat + (size_t)idx * DD) : (pool + (size_t)(idx - NU) * DD);
    *(float4*)(out + (size_t)ut * DD + c) = *(const float4*)(src + c);
}

// ------------------------------------------------ WMMA fp32 GEMM  out = A @ W (+bias)
// Block: 256 threads = 8 waves. Each block covers a 16-row stripe, each wave a
// 16x16 output tile (wave w -> cols [16w,16w+16)).  K-loop of 32
// V_WMMA_F32_16X16X4_F32 ops.  A staged in LDS; W (64KB) stays hot in L2/L1.
template <bool BIAS>
__global__ void __launch_bounds__(256) k_gemm(const float* __restrict__ A, const float* __restrict__ W,
                                              const float* __restrict__ bias, float* __restrict__ out,
                                              int M) {
    __shared__ float At[16][DD + 4];
    int r0 = blockIdx.x * 16;
    for (int i = threadIdx.x; i < 16 * 32; i += 256) {
        int r = i >> 5;
        int c = (i & 31) << 2;
        int row = r0 + r; if (row >= M) row = M - 1;
        float4 v = *(const float4*)(A + (size_t)row * DD + c);
        At[r][c] = v.x; At[r][c + 1] = v.y; At[r][c + 2] = v.z; At[r][c + 3] = v.w;
    }
    __syncthreads();

    int wave = threadIdx.x >> 5;
    int lane = threadIdx.x & 31;
    int half = lane >> 4;     // lanes 16-31 carry K+2/K+3 (A) and M+8 (C/D)
    int lr   = lane & 15;
    int n0   = wave * 16;
    int col  = n0 + lr;

    v8f c8 = {};
#pragma unroll 4
    for (int kk = 0; kk < DD; kk += 4) {
        v2f a, b;
        a.x = At[lr][kk + half * 2];
        a.y = At[lr][kk + half * 2 + 1];
        b.x = W[(size_t)(kk + half * 2) * DD + col];
        b.y = W[(size_t)(kk + half * 2 + 1) * DD + col];
        c8 = __builtin_amdgcn_wmma_f32_16x16x4_f32(false, a, false, b, (short)0, c8, false, false);
    }

    float bv = BIAS ? bias[col] : 0.f;
    int rowbase = r0 + half * 8;
#pragma unroll
    for (int i = 0; i < 8; i++) {
        int row = rowbase + i;
        if (row < M) out[(size_t)row * DD + col] = c8[i] + bv;
    }
}

// -------------------------- fusion score: score[k][n] += w2 . tanh(f_k[n] @ W1 + b1)
__global__ void __launch_bounds__(256) k_score(const float* __restrict__ sf, const float* __restrict__ df,
                                               const float* __restrict__ W1, const float* __restrict__ b1,
                                               const float* __restrict__ w2, float* __restrict__ score) {
    __shared__ float At[16][DD + 4];
    int k = blockIdx.y;
    const float* A = (k == 0) ? sf : df;
    int r0 = blockIdx.x * 16;
    for (int i = threadIdx.x; i < 16 * 32; i += 256) {
        int r = i >> 5;
        int c = (i & 31) << 2;
        float4 v = *(const float4*)(A + (size_t)(r0 + r) * DD + c);
        At[r][c] = v.x; At[r][c + 1] = v.y; At[r][c + 2] = v.z; At[r][c + 3] = v.w;
    }
    __syncthreads();

    int wave = threadIdx.x >> 5;
    int lane = threadIdx.x & 31;
    int half = lane >> 4;
    int lr   = lane & 15;
    int col  = wave * 16 + lr;

    v8f c8 = {};
#pragma unroll 4
    for (int kk = 0; kk < DD; kk += 4) {
        v2f a, b;
        a.x = At[lr][kk + half * 2];
        a.y = At[lr][kk + half * 2 + 1];
        b.x = W1[(size_t)(kk + half * 2) * DD + col];
        b.y = W1[(size_t)(kk + half * 2 + 1) * DD + col];
        c8 = __builtin_amdgcn_wmma_f32_16x16x4_f32(false, a, false, b, (short)0, c8, false, false);
    }

    float bb = b1[col], ww = w2[col];
    float p[8];
#pragma unroll
    for (int i = 0; i < 8; i++) p[i] = tanhf(c8[i] + bb) * ww;
    // reduce the 16 lanes of each half (VGPR i is row i / i+8 in halves 0/1)
#pragma unroll
    for (int m = 8; m >= 1; m >>= 1)
#pragma unroll
        for (int i = 0; i < 8; i++) p[i] += __shfl_xor(p[i], m, 32);
    if (lr == 0) {
#pragma unroll
        for (int i = 0; i < 8; i++)
            atomicAdd(&score[(size_t)k * NV + r0 + half * 8 + i], p[i]);
    }
}

// fused = softmax(score)_0 * sf + softmax(score)_1 * df   (b2 cancels in 2-way softmax)
__global__ void k_softmax_fuse(const float* __restrict__ sf, const float* __restrict__ df,
                               const float* __restrict__ score, float* __restrict__ fused) {
    long long g = (long long)blockIdx.x * blockDim.x + threadIdx.x;
    int n = (int)(g >> 5);
    int c = (int)(g & 31) << 2;
    if (n >= NV) return;
    float s0 = score[n], s1 = score[NV + n];
    float m = fmaxf(s0, s1);
    float e0 = __expf(s0 - m), e1 = __expf(s1 - m);
    float inv = 1.f / (e0 + e1);
    e0 *= inv; e1 *= inv;
    float4 a = *(const float4*)(sf + (size_t)n * DD + c);
    float4 b = *(const float4*)(df + (size_t)n * DD + c);
    float4 o;
    o.x = e0 * a.x + e1 * b.x; o.y = e0 * a.y + e1 * b.y;
    o.z = e0 * a.z + e1 * b.z; o.w = e0 * a.w + e1 * b.w;
    *(float4*)(fused + (size_t)n * DD + c) = o;
}

// ----------------------------------- SpMM scatter: out[rows[e]] += vals[e]*x[cols[e]]
// 32 threads per nnz, 4 columns each; f32 atomics resolve in L2.
__global__ void k_spmm(const int* __restrict__ rows, const int* __restrict__ cols,
                       const float* __restrict__ vals, const float* __restrict__ x,
                       float* __restrict__ out) {
    long long g = (long long)blockIdx.x * blockDim.x + threadIdx.x;
    int e = (int)(g >> 5);
    int c = (int)(g & 31) << 2;
    if (e >= NNZ) return;
    int r = rows[e];
    int cl = cols[e];
    float v = vals[e];
    float4 xv = *(const float4*)(x + (size_t)cl * DD + c);
    float* o = out + (size_t)r * DD + c;
    atomicAdd(o + 0, v * xv.x);
    atomicAdd(o + 1, v * xv.y);
    atomicAdd(o + 2, v * xv.z);
    atomicAdd(o + 3, v * xv.w);
}

// ================================================================= launcher
extern "C" void kernel_launch(void* const* d_in, const int* in_sizes, int n_in,
                              void* d_out, int out_size, void* d_ws, size_t ws_size,
                              hipStream_t stream) {
    const float* stat     = (const float*)d_in[0];
    const int*   guid_v   = (const int*)d_in[1];
    const int*   trace_v  = (const int*)d_in[2];
    const int*   trace_u  = (const int*)d_in[3];
    const int*   c1_rows  = (const int*)d_in[4];
    const int*   c1_cols  = (const int*)d_in[5];
    const float* c1_vals  = (const float*)d_in[6];
    const int*   c2_rows  = (const int*)d_in[7];
    const int*   c2_cols  = (const int*)d_in[8];
    const float* c2_vals  = (const float*)d_in[9];
    const int*   lg_rows  = (const int*)d_in[10];
    const int*   lg_cols  = (const int*)d_in[11];
    const float* lg_vals  = (const float*)d_in[12];
    const float* W_hgcn   = (const float*)d_in[13];
    const float* lg_w     = (const float*)d_in[14];
    const float* lg_b     = (const float*)d_in[15];
    const float* fw1      = (const float*)d_in[16];
    const float* fb1      = (const float*)d_in[17];
    const float* fw2      = (const float*)d_in[18];

    // ---- workspace layout (floats) ----
    float* ws = (float*)d_ws;
    size_t off = 0;
    float* xpool   = ws + off; off += (size_t)T_STEPS * NV * DD;  // dyn_v extension
    float* accpool = ws + off; off += (size_t)T_STEPS * NE * DD;  // dyn_u extension
    float* wsA = ws + off; off += (size_t)NV * DD;                // sf
    float* wsB = ws + off; off += (size_t)NV * DD;                // df
    float* wsC = ws + off; off += (size_t)NV * DD;
    float* wsD = ws + off; off += (size_t)NV * DD;
    float* wsE = ws + off; off += (size_t)NV * DD;
    float* score = ws + off; off += (size_t)2 * NV;
    if (ws_size < off * sizeof(float)) return;

    const long long rowN4   = (long long)NV * DD / 4;      // per [NV,128] buffer
    const int gEl   = cdiv(rowN4, 256);                    // elementwise over NV*128
    const int gRow  = cdiv((long long)NV * 32, 256);       // (n, col/4) threads
    const int gSp   = cdiv((long long)NNZ * 32, 256);      // spmm
    const int gTile = NV / 16;                             // 16-row GEMM stripes
    const int gFin  = cdiv((long long)NU * T_STEPS * 32, 256);

    for (int t = 0; t < T_STEPS; t++) {
        const int*   gv  = guid_v + (size_t)t * NV;
        const int*   r1  = c1_rows + (size_t)t * NNZ;
        const int*   cc1 = c1_cols + (size_t)t * NNZ;
        const float* v1  = c1_vals + (size_t)t * NNZ;
        const int*   r2  = c2_rows + (size_t)t * NNZ;
        const int*   cc2 = c2_cols + (size_t)t * NNZ;
        const float* v2  = c2_vals + (size_t)t * NNZ;
        const int*   rl  = lg_rows + (size_t)t * NNZ;
        const int*   ccl = lg_cols + (size_t)t * NNZ;
        const float* vl  = lg_vals + (size_t)t * NNZ;
        float* xcur = xpool + (size_t)t * NV * DD;
        float* acc  = accpool + (size_t)t * NE * DD;

        // -- gated fusion --
        k_gather_sfdf<<<gRow, 256, 0, stream>>>(stat, gv, trace_v, xpool, wsA, wsB, t);
        k_zero<<<cdiv(2 * NV / 4, 256), 256, 0, stream>>>(score, 2 * NV / 4);
        dim3 gs(gTile, 2);
        k_score<<<gs, 256, 0, stream>>>(wsA, wsB, fw1 + (size_t)t * DD * DD,
                                        fb1 + (size_t)t * DD, fw2 + (size_t)t * DD, score);
        k_softmax_fuse<<<gRow, 256, 0, stream>>>(wsA, wsB, score, wsC);

        // -- HGCN chain: 3 x (GEMM -> SpMM(coef1) -> ReLU) --
        const float* W0 = W_hgcn + ((size_t)t * (NHGCN + 1) + 0) * DD * DD;
        const float* W1 = W_hgcn + ((size_t)t * (NHGCN + 1) + 1) * DD * DD;
        const float* W2 = W_hgcn + ((size_t)t * (NHGCN + 1) + 2) * DD * DD;

        k_gemm<false><<<gTile, 256, 0, stream>>>(wsC, W0, nullptr, wsD, NV);
        k_zero<<<gEl, 256, 0, stream>>>(wsE, rowN4);
        k_spmm<<<gSp, 256, 0, stream>>>(r1, cc1, v1, wsD, wsE);
        k_relu<<<gEl, 256, 0, stream>>>(wsE, rowN4);

        k_gemm<false><<<gTile, 256, 0, stream>>>(wsE, W1, nullptr, wsD, NV);
        k_zero<<<gEl, 256, 0, stream>>>(wsC, rowN4);
        k_spmm<<<gSp, 256, 0, stream>>>(r1, cc1, v1, wsD, wsC);
        k_relu<<<gEl, 256, 0, stream>>>(wsC, rowN4);

        k_gemm<false><<<gTile, 256, 0, stream>>>(wsC, W2, nullptr, wsD, NV);
        k_zero<<<gEl, 256, 0, stream>>>(xcur, rowN4);
        k_spmm<<<gSp, 256, 0, stream>>>(r1, cc1, v1, wsD, xcur);
        k_relu<<<gEl, 256, 0, stream>>>(xcur, rowN4);

        // -- y = relu(spmm(coef2, x)) --
        k_zero<<<gEl, 256, 0, stream>>>(wsE, rowN4);
        k_spmm<<<gSp, 256, 0, stream>>>(r2, cc2, v2, xcur, wsE);
        k_relu<<<gEl, 256, 0, stream>>>(wsE, rowN4);

        // -- LineConv: acc = y; 2x (inp = spmm(L, inp@Wl + bl); acc += inp) --
        k_copy<<<gEl, 256, 0, stream>>>(acc, wsE, rowN4);

        const float* Wl0 = lg_w + ((size_t)t * NLG + 0) * DD * DD;
        const float* bl0 = lg_b + ((size_t)t * NLG + 0) * DD;
        k_gemm<true><<<gTile, 256, 0, stream>>>(wsE, Wl0, bl0, wsD, NE);
        k_zero<<<gEl, 256, 0, stream>>>(wsC, rowN4);
        k_spmm<<<gSp, 256, 0, stream>>>(rl, ccl, vl, wsD, wsC);
        k_add<<<gEl, 256, 0, stream>>>(acc, wsC, rowN4);

        const float* Wl1 = lg_w + ((size_t)t * NLG + 1) * DD * DD;
        const float* bl1 = lg_b + ((size_t)t * NLG + 1) * DD;
        k_gemm<true><<<gTile, 256, 0, stream>>>(wsC, Wl1, bl1, wsD, NE);
        k_zero<<<gEl, 256, 0, stream>>>(wsE, rowN4);
        k_spmm<<<gSp, 256, 0, stream>>>(rl, ccl, vl, wsD, wsE);
        k_add<<<gEl, 256, 0, stream>>>(acc, wsE, rowN4);
    }

    // -- final trace gathers: out = [final_u | final_v] --
    float* out_u = (float*)d_out;
    float* out_v = out_u + (size_t)NU * T_STEPS * DD;
    k_final_gather<<<gFin, 256, 0, stream>>>(trace_u, stat, accpool, out_u);
    k_final_gather<<<gFin, 256, 0, stream>>>(trace_v, stat, xpool, out_v);
}